// GATv2Layer_84447646974220
// MI455X (gfx1250) — compile-verified
//
#include <hip/hip_runtime.h>
#include <hip/hip_bf16.h>
#include <math.h>

// Problem constants (match reference)
#define NN      1024
#define INDIM   128
#define OUTDIM  128
#define NHEAD   4
#define HDIM    32
#define NEG_SLOPE 0.2f
#define MASK_VAL  (-1e9f)
#define LN_EPS    1e-5f

typedef __attribute__((ext_vector_type(2))) float v2f;
typedef __attribute__((ext_vector_type(8))) float v8f;

// ---------------------------------------------------------------------------
// Kernel 1: projections  Wlh = h@W_l, Wrh = h@W_r, Wv = h@W_v
// One wave32 computes one 16x16 f32 output tile with V_WMMA_F32_16X16X4_F32.
// Tiles: 3 matrices * (1024/16) * (128/16) = 3*64*8 = 1536. 4 waves/block ->
// 384 blocks, exact cover (EXEC all ones for every WMMA, as the ISA requires).
// ---------------------------------------------------------------------------
__global__ __launch_bounds__(128)
void gat_proj_wmma(const float* __restrict__ h,
                   const float* __restrict__ Wl,
                   const float* __restrict__ Wr,
                   const float* __restrict__ Wv,
                   float* __restrict__ proj /* [3][N][128] */) {
    const int lane = threadIdx.x & 31;
    const int wave = threadIdx.x >> 5;
    const int tile = blockIdx.x * 4 + wave;          // 0..1535
    const int mat  = tile / 512;                     // which matrix
    const int t    = tile % 512;
    const int mtile = t >> 3;                        // 0..63 (row tile)
    const int ntile = t & 7;                         // 0..7  (col tile)

    const float* W = (mat == 0) ? Wl : (mat == 1) ? Wr : Wv;
    float* O = proj + (size_t)mat * NN * OUTDIM;

    const int lhalf = lane >> 4;                     // 0 or 1
    const int lmod  = lane & 15;
    const int row   = mtile * 16 + lmod;             // A-matrix M
    const int col   = ntile * 16 + lmod;             // B/C/D N
    const int kh    = lhalf * 2;                     // K sub-offset per lane half

    v8f acc = {0.f, 0.f, 0.f, 0.f, 0.f, 0.f, 0.f, 0.f};

    for (int k0 = 0; k0 < INDIM; k0 += 4) {
        v2f av, bv;
        // A 16x4: VGPR0 -> K = k0+kh, VGPR1 -> K = k0+kh+1  (M = lane&15)
        av.x = h[row * INDIM + k0 + kh + 0];
        av.y = h[row * INDIM + k0 + kh + 1];
        // B 4x16: VGPR g -> K = k0 + g + 2*(lane>=16), N = lane&15
        bv.x = W[(k0 + kh + 0) * OUTDIM + col];
        bv.y = W[(k0 + kh + 1) * OUTDIM + col];
        acc = __builtin_amdgcn_wmma_f32_16x16x4_f32(
            /*neg_a=*/false, av, /*neg_b=*/false, bv,
            /*c_mod=*/(short)0, acc, /*reuse_a=*/false, /*reuse_b=*/false);
    }

    // C/D 16x16 f32: VGPR g holds M = g + 8*(lane>=16), N = lane&15
    const int mbase = mtile * 16 + lhalf * 8;
#pragma unroll
    for (int g = 0; g < 8; ++g) {
        O[(mbase + g) * OUTDIM + col] = acc[g];
    }
}

// ---------------------------------------------------------------------------
// Kernel 2: rank-1 precompute  La[i,h] = <a, Wlh[i,h,:]>, Ra[j,h] = <a, Wrh[j,h,:]>
// ---------------------------------------------------------------------------
__global__ __launch_bounds__(256)
void gat_rank1(const float* __restrict__ proj,   // [3][N][128]
               const float* __restrict__ a_vec,  // [32]
               float* __restrict__ La,           // [N*H]
               float* __restrict__ Ra) {         // [N*H]
    int t = blockIdx.x * 256 + threadIdx.x;      // 0..8191
    int which = t >> 12;                         // 0 = L, 1 = R
    int idx   = t & 4095;                        // i*H + h
    int i  = idx >> 2;
    int hh = idx & 3;
    const float* src = proj + (size_t)which * NN * OUTDIM;
    float s = 0.f;
#pragma unroll
    for (int d = 0; d < HDIM; ++d)
        s = fmaf(a_vec[d], src[i * OUTDIM + hh * HDIM + d], s);
    (which ? Ra : La)[idx] = s;
}

// ---------------------------------------------------------------------------
// Kernel 3: fused per-row attention + softmax + aggregation + LayerNorm + ReLU
// One block (128 threads = 4 wave32) per node i. Scores for the whole row
// (1024 j x 4 heads = 16 KB) live in LDS (320 KB/WGP -> high occupancy).
//
// e[i,j,h] = 0.6*(La[i,h]+Ra[j,h]) + 0.4 * sum_d a_d * |Wlh[i,h,d]+Wrh[j,h,d]|
// (identity: leaky_relu_{0.2}(x) = 0.6x + 0.4|x|; |x| is a free VALU modifier)
// ---------------------------------------------------------------------------
__global__ __launch_bounds__(128)
void gat_attn_row(const float* __restrict__ proj,   // [3][N][128]
                  const int*   __restrict__ adj,    // [N][N]
                  const float* __restrict__ a_vec,  // [32]
                  const float* __restrict__ La,     // [N*H]
                  const float* __restrict__ Ra,     // [N*H]
                  const float* __restrict__ ln_g,
                  const float* __restrict__ ln_b,
                  float* __restrict__ out) {        // [N][128]
    __shared__ float lds_e[NN * NHEAD];     // 16 KB: scores, then exp(e-max)
    __shared__ float lds_wl[OUTDIM];        // Wlh row i
    __shared__ float lds_a[HDIM];           // attention vector
    __shared__ float lds_red[32 * NHEAD];   // per-chunk partials
    __shared__ float lds_hmax[NHEAD];
    __shared__ float lds_hsum[NHEAD];
    __shared__ float lds_stat[2];           // mu, inv-sigma

    const int i = blockIdx.x;
    const int t = threadIdx.x;              // 0..127

    const float* Wlh = proj;
    const float* Wrh = proj + (size_t)NN * OUTDIM;
    const float* Wv  = proj + (size_t)2 * NN * OUTDIM;

    lds_wl[t] = Wlh[i * OUTDIM + t];
    if (t < HDIM) lds_a[t] = a_vec[t];
    __syncthreads();

    float la0 = La[i * NHEAD + 0], la1 = La[i * NHEAD + 1];
    float la2 = La[i * NHEAD + 2], la3 = La[i * NHEAD + 3];

    // ---- Pass 1: scores into LDS (each thread owns 8 j values) ----
    for (int j = t; j < NN; j += 128) {
        if (adj[i * NN + j] != 0) {
            const float* wr = Wrh + j * OUTDIM;
            float e0, e1, e2, e3;
            {
                float s = 0.f;
#pragma unroll
                for (int d = 0; d < HDIM; ++d)
                    s = fmaf(lds_a[d], fabsf(lds_wl[0 * HDIM + d] + wr[0 * HDIM + d]), s);
                e0 = 0.6f * (la0 + Ra[j * NHEAD + 0]) + 0.4f * s;
            }
            {
                float s = 0.f;
#pragma unroll
                for (int d = 0; d < HDIM; ++d)
                    s = fmaf(lds_a[d], fabsf(lds_wl[1 * HDIM + d] + wr[1 * HDIM + d]), s);
                e1 = 0.6f * (la1 + Ra[j * NHEAD + 1]) + 0.4f * s;
            }
            {
                float s = 0.f;
#pragma unroll
                for (int d = 0; d < HDIM; ++d)
                    s = fmaf(lds_a[d], fabsf(lds_wl[2 * HDIM + d] + wr[2 * HDIM + d]), s);
                e2 = 0.6f * (la2 + Ra[j * NHEAD + 2]) + 0.4f * s;
            }
            {
                float s = 0.f;
#pragma unroll
                for (int d = 0; d < HDIM; ++d)
                    s = fmaf(lds_a[d], fabsf(lds_wl[3 * HDIM + d] + wr[3 * HDIM + d]), s);
                e3 = 0.6f * (la3 + Ra[j * NHEAD + 3]) + 0.4f * s;
            }
            lds_e[j * NHEAD + 0] = e0;
            lds_e[j * NHEAD + 1] = e1;
            lds_e[j * NHEAD + 2] = e2;
            lds_e[j * NHEAD + 3] = e3;
        } else {
            lds_e[j * NHEAD + 0] = MASK_VAL;
            lds_e[j * NHEAD + 1] = MASK_VAL;
            lds_e[j * NHEAD + 2] = MASK_VAL;
            lds_e[j * NHEAD + 3] = MASK_VAL;
        }
    }
    __syncthreads();

    // ---- Pass 2: per-head softmax (max, then exp + sum), in LDS ----
    const int h4 = t & 3;          // head this thread reduces
    const int ch = t >> 2;         // chunk of 32 j values
    {
        float m = -INFINITY;
        for (int j = ch * 32; j < ch * 32 + 32; ++j)
            m = fmaxf(m, lds_e[j * NHEAD + h4]);
        lds_red[ch * NHEAD + h4] = m;
    }
    __syncthreads();
    if (t < NHEAD) {
        float m = -INFINITY;
        for (int c = 0; c < 32; ++c) m = fmaxf(m, lds_red[c * NHEAD + t]);
        lds_hmax[t] = m;
    }
    __syncthreads();
    {
        const float hm = lds_hmax[h4];
        float s = 0.f;
        for (int j = ch * 32; j < ch * 32 + 32; ++j) {
            float p = __expf(lds_e[j * NHEAD + h4] - hm);
            lds_e[j * NHEAD + h4] = p;
            s += p;
        }
        lds_red[ch * NHEAD + h4] = s;
    }
    __syncthreads();
    if (t < NHEAD) {
        float s = 0.f;
        for (int c = 0; c < 32; ++c) s += lds_red[c * NHEAD + t];
        lds_hsum[t] = s;
    }
    __syncthreads();

    // ---- Pass 3: aggregation. Thread t owns output column (h = t>>5, d = t&31).
    // lds_e read is a wave-wide broadcast; Wv read is fully coalesced (128B/j/wave).
    const int hh = t >> 5;
    const int dd = t & 31;
    float acc = 0.f;
    for (int j = 0; j < NN; ++j)
        acc = fmaf(lds_e[j * NHEAD + hh], Wv[j * OUTDIM + hh * HDIM + dd], acc);
    const float x = acc / lds_hsum[hh];

    // ---- Pass 4: LayerNorm over the 128 columns of row i, then ReLU ----
    __syncthreads();                // lds_e free now
    lds_e[t] = x;
    lds_e[128 + t] = x * x;
    __syncthreads();
    if (t == 0) {
        float s1 = 0.f, s2 = 0.f;
        for (int k = 0; k < 128; ++k) { s1 += lds_e[k]; s2 += lds_e[128 + k]; }
        float mu  = s1 * (1.0f / 128.0f);
        float var = s2 * (1.0f / 128.0f) - mu * mu;
        lds_stat[0] = mu;
        lds_stat[1] = rsqrtf(var + LN_EPS);
    }
    __syncthreads();
    float y = (x - lds_stat[0]) * lds_stat[1] * ln_g[t] + ln_b[t];
    out[i * OUTDIM + t] = fmaxf(y, 0.f);
}

// ---------------------------------------------------------------------------
// Launch
// ---------------------------------------------------------------------------
extern "C" void kernel_launch(void* const* d_in, const int* in_sizes, int n_in,
                              void* d_out, int out_size, void* d_ws, size_t ws_size,
                              hipStream_t stream) {
    (void)in_sizes; (void)n_in; (void)out_size; (void)ws_size;

    const float* h    = (const float*)d_in[0];
    const int*   adj  = (const int*)  d_in[1];
    const float* Wl   = (const float*)d_in[2];
    const float* Wr   = (const float*)d_in[3];
    const float* Wv   = (const float*)d_in[4];
    const float* avec = (const float*)d_in[5];
    const float* ln_g = (const float*)d_in[6];
    const float* ln_b = (const float*)d_in[7];
    float* out = (float*)d_out;

    // Workspace layout (floats):
    //   proj[3][N][128] : 393216   (Wlh, Wrh, Wv)
    //   La[N*H]         : 4096
    //   Ra[N*H]         : 4096
    float* proj = (float*)d_ws;
    float* La   = proj + (size_t)3 * NN * OUTDIM;
    float* Ra   = La + (size_t)NN * NHEAD;

    // 1) projections via f32 WMMA: 1536 tiles, 4 waves/block -> 384 blocks
    gat_proj_wmma<<<384, 128, 0, stream>>>(h, Wl, Wr, Wv, proj);

    // 2) rank-1 dot products with `a`: 8192 items
    gat_rank1<<<32, 256, 0, stream>>>(proj, avec, La, Ra);

    // 3) fused attention + softmax + aggregation + LN + ReLU: one block per row
    gat_attn_row<<<NN, 128, 0, stream>>>(proj, adj, avec, La, Ra, ln_g, ln_b, out);
}